// DGPE_ODE_RELAXATION_2723009266046
// MI455X (gfx1250) — compile-verified
//
#include <hip/hip_runtime.h>

// DGPE ODE RHS on a periodic 192^3 lattice — bandwidth-roofline version.
//   44 B/site compulsory traffic @ 23.3 TB/s  =>  ~13.4 us floor.
// Round-1 kernel was issue-bound (~142 inst/site). This version amortizes 4
// sites per thread with b128 accesses everywhere:
//   - neighbor indices computed analytically (nn_id* arrays never loaded)
//   - k-rows staged to LDS via gfx1250 async DMA (global_load_async_to_lds_b128,
//     completion via s_wait_asynccnt / ASYNCcnt)
//   - single-use coefficient streams + output: non-temporal (L2 kept for x/p)

typedef float v4f __attribute__((ext_vector_type(4)));

#define LDIM 192
#define NPTS (LDIM * LDIM * LDIM)

__global__ __launch_bounds__(192) void dgpe_rhs_kernel(
    const float* __restrict__ y,      // [2N] : x | p
    const float* __restrict__ Jf,     // [N]
    const float* __restrict__ aniso,  // [N]
    const float* __restrict__ gam,    // [N]
    const float* __restrict__ hdx,    // [N]
    const float* __restrict__ hdy,    // [N]
    const float* __restrict__ bet,    // [N]
    const float* __restrict__ edis,   // [N]
    float* __restrict__ out)          // [2N] : dx | dp
{
    // 4 j-rows of x and p staged per block (6 KB LDS).
    __shared__ __align__(16) float sx[4][LDIM];
    __shared__ __align__(16) float sp[4][LDIM];

    const int q  = threadIdx.x;            // k-quad index, 0..47
    const int jj = threadIdx.y;            // row within block's j-tile, 0..3
    const int i  = blockIdx.y;             // slow axis (stride L*L)
    const int j  = blockIdx.x * 4 + jj;    // middle axis (stride L)
    const int k0 = q << 2;                 // fast axis (stride 1), 16B aligned
    const int n  = (i * LDIM + j) * LDIM + k0;

    const float* __restrict__ xg = y;
    const float* __restrict__ pg = y + NPTS;

    // ---- async DMA this quad of x and p into LDS (gfx1250 ASYNCcnt path) ----
    {
        unsigned lx = (unsigned)(size_t)&sx[jj][k0];   // low 32b = LDS offset
        unsigned lp = (unsigned)(size_t)&sp[jj][k0];
        unsigned long long gx = (unsigned long long)(const void*)(xg + n);
        unsigned long long gp = (unsigned long long)(const void*)(pg + n);
        asm volatile("global_load_async_to_lds_b128 %0, %1, off"
                     :: "v"(lx), "v"(gx) : "memory");
        asm volatile("global_load_async_to_lds_b128 %0, %1, off"
                     :: "v"(lp), "v"(gp) : "memory");
    }

    // ---- periodic neighbor rows (analytic, no gather-index traffic) ----
    const int im = (i == 0)        ? (LDIM - 1) : (i - 1);
    const int ip = (i == LDIM - 1) ? 0          : (i + 1);
    const int jm = (j == 0)        ? (LDIM - 1) : (j - 1);
    const int jp = (j == LDIM - 1) ? 0          : (j + 1);

    const int n_im = (im * LDIM + j ) * LDIM + k0;
    const int n_ip = (ip * LDIM + j ) * LDIM + k0;
    const int n_jm = (i  * LDIM + jm) * LDIM + k0;
    const int n_jp = (i  * LDIM + jp) * LDIM + k0;

    // ---- x/y-direction neighbor quads from global (L2-resident, b128) ----
    const v4f x_im = *(const v4f*)(xg + n_im);
    const v4f x_ip = *(const v4f*)(xg + n_ip);
    const v4f x_jm = *(const v4f*)(xg + n_jm);
    const v4f x_jp = *(const v4f*)(xg + n_jp);
    const v4f p_im = *(const v4f*)(pg + n_im);
    const v4f p_ip = *(const v4f*)(pg + n_ip);
    const v4f p_jm = *(const v4f*)(pg + n_jm);
    const v4f p_jp = *(const v4f*)(pg + n_jp);

    // ---- single-use coefficient streams: non-temporal b128 ----
    const v4f Jc = __builtin_nontemporal_load((const v4f*)(Jf    + n));
    const v4f an = __builtin_nontemporal_load((const v4f*)(aniso + n));
    const v4f ga = __builtin_nontemporal_load((const v4f*)(gam   + n));
    const v4f hx = __builtin_nontemporal_load((const v4f*)(hdx   + n));
    const v4f hy = __builtin_nontemporal_load((const v4f*)(hdy   + n));
    const v4f bt = __builtin_nontemporal_load((const v4f*)(bet   + n));
    const v4f ed = __builtin_nontemporal_load((const v4f*)(edis  + n));

    // Async LDS fills of this wave complete, then block-wide visibility.
    asm volatile("s_wait_asynccnt 0" ::: "memory");
    __syncthreads();

    // ---- k-direction stencil: quad + two wrap-edge scalars from LDS ----
    const int ke_m = (k0 == 0)        ? (LDIM - 1) : (k0 - 1);
    const int ke_p = (k0 == LDIM - 4) ? 0          : (k0 + 4);

    const v4f   xq  = *(const v4f*)&sx[jj][k0];
    const v4f   pq  = *(const v4f*)&sp[jj][k0];
    const float xlo = sx[jj][ke_m], xhi = sx[jj][ke_p];
    const float plo = sp[jj][ke_m], phi = sp[jj][ke_p];

    const v4f x_km = { xlo,  xq.x, xq.y, xq.z };
    const v4f x_kp = { xq.y, xq.z, xq.w, xhi  };
    const v4f p_km = { plo,  pq.x, pq.y, pq.z };
    const v4f p_kp = { pq.y, pq.z, pq.w, phi  };

    // stencil(v) = J * (v_x- + v_x+ + v_y- + v_y+ + aniso*(v_z- + v_z+))
    const v4f xL = Jc * (x_im + x_ip + x_jm + x_jp + an * (x_km + x_kp));
    const v4f yL = Jc * (p_im + p_ip + p_jm + p_jp + an * (p_km + p_kp));

    const v4f r2    = xq * xq + pq * pq;
    const v4f cross = xL * pq - yL * xq;

    const v4f dx =  ga * pq * cross + ed * pq - yL + hy + bt * r2 * pq;
    const v4f dp = -(ga * xq * cross) - ed * xq + xL - hx - bt * r2 * xq;

    // Streaming output: non-temporal b128 stores.
    __builtin_nontemporal_store(dx, (v4f*)(out + n));
    __builtin_nontemporal_store(dp, (v4f*)(out + NPTS + n));
}

extern "C" void kernel_launch(void* const* d_in, const int* in_sizes, int n_in,
                              void* d_out, int out_size, void* d_ws, size_t ws_size,
                              hipStream_t stream) {
    // setup_inputs order:
    // 0:t 1:y 2:J 3:anisotropy 4:gamma 5:h_dis_x 6:h_dis_y 7:beta 8:e_disorder
    // 9..14: nn_id* (unused -- neighbors computed analytically)
    (void)in_sizes; (void)n_in; (void)out_size; (void)d_ws; (void)ws_size;

    const float* y   = (const float*)d_in[1];
    const float* J   = (const float*)d_in[2];
    const float* an  = (const float*)d_in[3];
    const float* ga  = (const float*)d_in[4];
    const float* hdx = (const float*)d_in[5];
    const float* hdy = (const float*)d_in[6];
    const float* bt  = (const float*)d_in[7];
    const float* ed  = (const float*)d_in[8];

    dim3 grid(LDIM / 4, LDIM, 1);   // (j-tiles, i)
    dim3 block(LDIM / 4, 4, 1);     // 48 k-quads x 4 j-rows = 192 thr / 6 waves
    dgpe_rhs_kernel<<<grid, block, 0, stream>>>(y, J, an, ga, hdx, hdy, bt, ed,
                                                (float*)d_out);
}